// KNN_40948218200127
// MI455X (gfx1250) — compile-verified
//
#include <hip/hip_runtime.h>

// KNN post-processing (RangeNet++-style) for MI455X / gfx1250.
// Gather/selection bound; working set is fully L2-resident (proj_range 512 KB,
// proj_argmax 10 MB << 192 MB L2). No shared-operand matmul structure ->
// WMMA would only inflate data movement; optimum is vectorized L2 gathers
// plus a minimal-VALU med3-based top-5 selection network.

typedef __attribute__((ext_vector_type(4))) float v4f;

namespace {
constexpr int   SEARCH = 5;
constexpr int   PADR   = 2;
constexpr int   CENTER = 12;
constexpr int   KNN_K  = 5;
constexpr float CUTOFF = 1.0f;
constexpr int   H      = 64;
constexpr int   W      = 2048;
constexpr int   C      = 20;

__device__ __forceinline__ int iclamp(int v, int lo, int hi) {
    return v < lo ? lo : (v > hi ? hi : v);
}

__device__ __forceinline__ unsigned umin2(unsigned a, unsigned b) {
    return a < b ? a : b;
}
__device__ __forceinline__ unsigned umax2(unsigned a, unsigned b) {
    return a > b ? a : b;
}
// Canonical med3 pattern -> backend emits a single v_med3_u32.
__device__ __forceinline__ unsigned umed3(unsigned a, unsigned b, unsigned c) {
    return umax2(umin2(a, b), umin2(umax2(a, b), c));
}

// Insert key k into the sorted (ascending) 5-list m, dropping the largest.
// m0' = min(m0,k); mi' = med3(m(i-1), mi, k) -- 1 v_min_u32 + 4 v_med3_u32,
// all reading OLD list values => depth-1, fully dual-issueable.
__device__ __forceinline__ void insert5(unsigned k, unsigned m[KNN_K]) {
    const unsigned n0 = umin2(m[0], k);
    const unsigned n1 = umed3(m[0], m[1], k);
    const unsigned n2 = umed3(m[1], m[2], k);
    const unsigned n3 = umed3(m[2], m[3], k);
    const unsigned n4 = umed3(m[3], m[4], k);
    m[0] = n0; m[1] = n1; m[2] = n2; m[3] = n3; m[4] = n4;
}
} // namespace

__global__ __launch_bounds__(256) void knn_post_kernel(
    const float* __restrict__ proj_range,    // [H,W]
    const float* __restrict__ unproj_range,  // [P]
    const float* __restrict__ proj_argmax,   // [H,W,C]
    const int*   __restrict__ px,            // [P]
    const int*   __restrict__ py,            // [P]
    float*       __restrict__ out,           // [P,C]
    int P)
{
    const int p = blockIdx.x * blockDim.x + threadIdx.x;
    if (p >= P) return;

    // 1 - normalized 5x5 gaussian (sigma=1), indexed by dy*dy+dx*dx.
    // Constant-folded to immediates by full unrolling.
    const float INVG_D2[9] = {
        0.83789718f,  // r2=0 (center; multiplies d==0 anyway)
        0.90167967f,  // r2=1
        0.94036570f,  // r2=2
        0.0f,
        0.97806177f,  // r2=4
        0.98669379f,  // r2=5
        0.0f, 0.0f,
        0.99703098f   // r2=8
    };

    const int   cx = px[p];
    const int   cy = py[p];
    const float ur = unproj_range[p];

    // The center row always contributes (d[CENTER]==0 <= CUTOFF): warm it up
    // while the 24 scattered range gathers + selection network run.
    __builtin_prefetch(proj_argmax + (cy * W + cx) * C, 0, 3);

    // ---- top-5 selection via monotonic u32 keys --------------------------
    // d >= 0, so IEEE bits order == numeric order. key = (bits & ~31) | j:
    // upper 27 bits = magnitude, low 5 bits = window index (unique keys,
    // lowest-index-first on ties, matching top_k).
    unsigned m[KNN_K];
    #pragma unroll
    for (int i = 0; i < KNN_K; ++i) m[i] = 0xFFFFFFFFu;

    #pragma unroll
    for (int ky = 0; ky < SEARCH; ++ky) {
        const int  y   = cy + ky - PADR;
        const bool yok = (unsigned)y < (unsigned)H;
        const int  yc  = iclamp(y, 0, H - 1);
        #pragma unroll
        for (int kx = 0; kx < SEARCH; ++kx) {
            const int j = ky * SEARCH + kx;
            if (j == CENTER) {             // rang[center]=ur -> d==0
                insert5((unsigned)CENTER, m);
                continue;
            }
            const int  x  = cx + kx - PADR;
            const bool ok = yok && ((unsigned)x < (unsigned)W);
            const int  xc = iclamp(x, 0, W - 1);
            float r = proj_range[yc * W + xc];       // clamped, always in-bounds
            r = ok ? r : 0.0f;                       // F.unfold zero padding
            if (r < 0.0f) r = __builtin_inff();
            const int dy = ky - PADR, dx = kx - PADR;
            const float d = fabsf(r - ur) * INVG_D2[dy * dy + dx * dx];
            insert5((__float_as_uint(d) & 0xFFFFFFE0u) | (unsigned)j, m);
        }
    }

    // ---- accumulate the (<= cutoff, in-bounds) winners -------------------
    float acc[C];
    #pragma unroll
    for (int c = 0; c < C; ++c) acc[c] = 0.0f;

    #pragma unroll
    for (int k = 0; k < KNN_K; ++k) {
        const unsigned mk = m[k];
        const int   bi = (int)(mk & 31u);
        const float bv = __uint_as_float(mk & 0xFFFFFFE0u);
        const int dy = bi / SEARCH;                  // strength-reduced
        const int dx = bi - dy * SEARCH;
        const int y  = cy + dy - PADR;
        const int x  = cx + dx - PADR;
        const bool ok = ((unsigned)y < (unsigned)H) && ((unsigned)x < (unsigned)W);

        // EXEC-masked gather: lanes failing the cutoff issue no VMEM requests
        // (with uniform-random ranges only ~3% of non-center neighbors pass).
        if (ok && (bv <= CUTOFF)) {
            const v4f* row = (const v4f*)(proj_argmax + (y * W + x) * C); // 16B aligned
            #pragma unroll
            for (int q = 0; q < C / 4; ++q) {
                const v4f v = row[q];                // global_load_b128
                acc[4 * q + 0] += v.x;
                acc[4 * q + 1] += v.y;
                acc[4 * q + 2] += v.z;
                acc[4 * q + 3] += v.w;
            }
        }
    }

    // ---- write-once output: non-temporal b128 stores, keep L2 for gathers --
    v4f* orow = (v4f*)(out + (size_t)p * C);         // p*80B -> 16B aligned
    #pragma unroll
    for (int q = 0; q < C / 4; ++q) {
        v4f v;
        v.x = acc[4 * q + 0];
        v.y = acc[4 * q + 1];
        v.z = acc[4 * q + 2];
        v.w = acc[4 * q + 3];
        __builtin_nontemporal_store(v, orow + q);
    }
}

extern "C" void kernel_launch(void* const* d_in, const int* in_sizes, int n_in,
                              void* d_out, int out_size, void* d_ws, size_t ws_size,
                              hipStream_t stream) {
    const float* proj_range   = (const float*)d_in[0];
    const float* unproj_range = (const float*)d_in[1];
    const float* proj_argmax  = (const float*)d_in[2];
    const int*   px           = (const int*)d_in[3];
    const int*   py           = (const int*)d_in[4];
    float*       out          = (float*)d_out;

    const int P = in_sizes[1];                 // unproj_range element count
    const int threads = 256;                   // 8 wave32 per workgroup
    const int blocks  = (P + threads - 1) / threads;

    knn_post_kernel<<<blocks, threads, 0, stream>>>(
        proj_range, unproj_range, proj_argmax, px, py, out, P);
}